// GINConvNet_33629593928256
// MI455X (gfx1250) — compile-verified
//
#include <hip/hip_runtime.h>

typedef __attribute__((ext_vector_type(2))) float v2f;
typedef __attribute__((ext_vector_type(8))) float v8f;

#define DIMF 31
#define PAD 32

__device__ __forceinline__ v8f wmma_f32(v2f a, v2f b, v8f c) {
  // V_WMMA_F32_16X16X4_F32 : D = A(16x4) x B(4x16) + C(16x16), all fp32
  return __builtin_amdgcn_wmma_f32_16x16x4_f32(
      /*neg_a=*/false, a, /*neg_b=*/false, b,
      /*c_mod=*/(short)0, c, /*reuse_a=*/false, /*reuse_b=*/false);
}

// ---------------- utility kernels ----------------

__global__ void k_zero(float* __restrict__ p, size_t n) {
  size_t i = (size_t)blockIdx.x * blockDim.x + threadIdx.x;
  size_t stride = (size_t)gridDim.x * blockDim.x;
  for (; i < n; i += stride) p[i] = 0.f;
}

__global__ void k_pad_copy(const float* __restrict__ x, float* __restrict__ xp, int N) {
  size_t i = (size_t)blockIdx.x * blockDim.x + threadIdx.x;
  int r = (int)(i >> 5), c = (int)(i & 31);
  if (r < N) xp[i] = (c < DIMF) ? x[(size_t)r * DIMF + c] : 0.f;
}

__global__ void k_add(const float* __restrict__ a, const float* __restrict__ b,
                      float* __restrict__ o, size_t n) {
  size_t i = (size_t)blockIdx.x * blockDim.x + threadIdx.x;
  if (i < n) o[i] = a[i] + b[i];
}

// ---------------- graph scatter: agg[dst] += x[src] ----------------

__global__ void k_scatter(const int* __restrict__ src, const int* __restrict__ dst,
                          const float* __restrict__ xp, float* __restrict__ agg, int E) {
  size_t i = (size_t)blockIdx.x * blockDim.x + threadIdx.x;
  int e = (int)(i >> 5), c = (int)(i & 31);
  if (e < E && c < DIMF) {
    float v = xp[(size_t)src[e] * PAD + c];
    atomicAdd(&agg[(size_t)dst[e] * PAD + c], v);
  }
}

// ---------------- graph pooling ----------------

__global__ void k_pool(const float* __restrict__ xp, const int* __restrict__ batch,
                       float* __restrict__ pooled, int N) {
  size_t i = (size_t)blockIdx.x * blockDim.x + threadIdx.x;
  int r = (int)(i >> 5), c = (int)(i & 31);
  if (r < N && c < DIMF)
    atomicAdd(&pooled[(size_t)batch[r] * PAD + c], xp[i]);
}

// ---------------- GIN MLP: h = relu( relu((x+agg)@W1+b1) @ W2 + b2 ) + BN stats ----------------

__global__ __launch_bounds__(128) void k_gin_mlp(
    const float* __restrict__ xp, const float* __restrict__ agg, float* __restrict__ h,
    const float* __restrict__ W1, const float* __restrict__ B1,
    const float* __restrict__ W2, const float* __restrict__ B2,
    float* __restrict__ stats, int N) {
  __shared__ float sW1[PAD][PAD];
  __shared__ float sW2[PAD][PAD];
  __shared__ float sIn[4][16][PAD];

  int tid = threadIdx.x;
  int wave = tid >> 5, lane = tid & 31;
  int half = lane >> 4, l16 = lane & 15;

  // stage 31x31 weights zero-padded to 32x32
  for (int i = tid; i < PAD * PAD; i += 128) {
    int r = i >> 5, c = i & 31;
    bool ok = (r < DIMF) && (c < DIMF);
    sW1[r][c] = ok ? W1[r * DIMF + c] : 0.f;
    sW2[r][c] = ok ? W2[r * DIMF + c] : 0.f;
  }
  __syncthreads();

  size_t tile = (size_t)blockIdx.x * 4 + wave;  // 16-row tile per wave
  size_t row0 = tile * 16;

  // load input tile (x + agg), lane = column
  for (int i = 0; i < 16; ++i) {
    size_t r = row0 + i;
    size_t rc = (r < (size_t)N) ? r : (size_t)(N - 1);
    float v = xp[rc * PAD + lane] + agg[rc * PAD + lane];
    if (r >= (size_t)N) v = 0.f;
    sIn[wave][i][lane] = v;
  }

  // ---- GEMM1: C[16x32] = A[16x32] @ W1[32x32], via 8 k-steps x 2 col-tiles
  v8f c0 = {}, c1 = {};
#pragma unroll
  for (int k = 0; k < PAD; k += 4) {
    v2f a, b0, b1v;
    a.x = sIn[wave][l16][k + 2 * half];
    a.y = sIn[wave][l16][k + 2 * half + 1];
    b0.x = sW1[k + half][l16];
    b0.y = sW1[k + 2 + half][l16];
    b1v.x = sW1[k + half][16 + l16];
    b1v.y = sW1[k + 2 + half][16 + l16];
    c0 = wmma_f32(a, b0, c0);
    c1 = wmma_f32(a, b1v, c1);
  }
  float bb0 = B1[l16];
  float bb1 = (16 + l16 < DIMF) ? B1[16 + l16] : 0.f;
#pragma unroll
  for (int i = 0; i < 8; ++i) {
    sIn[wave][i + 8 * half][l16] = fmaxf(c0[i] + bb0, 0.f);
    sIn[wave][i + 8 * half][16 + l16] = fmaxf(c1[i] + bb1, 0.f);
  }

  // ---- GEMM2 with W2
  v8f d0 = {}, d1 = {};
#pragma unroll
  for (int k = 0; k < PAD; k += 4) {
    v2f a, b0, b1v;
    a.x = sIn[wave][l16][k + 2 * half];
    a.y = sIn[wave][l16][k + 2 * half + 1];
    b0.x = sW2[k + half][l16];
    b0.y = sW2[k + 2 + half][l16];
    b1v.x = sW2[k + half][16 + l16];
    b1v.y = sW2[k + 2 + half][16 + l16];
    d0 = wmma_f32(a, b0, d0);
    d1 = wmma_f32(a, b1v, d1);
  }
  float cb0 = B2[l16];
  float cb1 = (16 + l16 < DIMF) ? B2[16 + l16] : 0.f;
  float s0 = 0.f, q0 = 0.f, s1 = 0.f, q1 = 0.f;
#pragma unroll
  for (int i = 0; i < 8; ++i) {
    size_t r = row0 + i + 8 * half;
    float v0 = fmaxf(d0[i] + cb0, 0.f);
    float v1 = (16 + l16 < DIMF) ? fmaxf(d1[i] + cb1, 0.f) : 0.f;
    if (r < (size_t)N) {
      h[r * PAD + l16] = v0;
      h[r * PAD + 16 + l16] = v1;
      s0 += v0; q0 += v0 * v0;
      s1 += v1; q1 += v1 * v1;
    }
  }
  // BN statistics: stats[0..31]=sum, stats[32..63]=sumsq
  atomicAdd(&stats[l16], s0);
  atomicAdd(&stats[32 + l16], q0);
  atomicAdd(&stats[16 + l16], s1);
  atomicAdd(&stats[48 + l16], q1);
}

__global__ void k_bn_finalize(const float* __restrict__ stats, float* __restrict__ meaninv,
                              float invN) {
  int f = threadIdx.x;  // 32 threads
  float m = stats[f] * invN;
  float v = stats[32 + f] * invN - m * m;
  meaninv[f] = m;
  meaninv[32 + f] = rsqrtf(v + 1e-5f);
}

__global__ void k_bn_apply(const float* __restrict__ h, const float* __restrict__ meaninv,
                           const float* __restrict__ gamma, const float* __restrict__ beta,
                           float* __restrict__ xo, size_t total) {
  size_t i = (size_t)blockIdx.x * blockDim.x + threadIdx.x;
  if (i >= total) return;
  int c = (int)(i & 31);
  float v = 0.f;
  if (c < DIMF)
    v = (h[i] - meaninv[c]) * meaninv[32 + c] * gamma[c] + beta[c];
  xo[i] = v;
}

// ---------------- generic fp32 WMMA GEMM: C[M,Nn] = act(A[M,lda] @ W[K,Nn] + bias) ----------------

__global__ __launch_bounds__(128) void k_gemm(
    const float* __restrict__ A, int lda, const float* __restrict__ W,
    const float* __restrict__ bias, float* __restrict__ C, int ldc,
    int M, int Nn, int K, int Kp, int relu) {
  int tid = threadIdx.x;
  int wave = tid >> 5, lane = tid & 31;
  int half = lane >> 4, l16 = lane & 15;
  int ntiles = Nn >> 4;
  long gtile = (long)blockIdx.x * 4 + wave;
  int mt = (int)(gtile / ntiles);
  int nt = (int)(gtile % ntiles);
  if (mt * 16 >= M) return;  // wave-uniform exit

  int n = nt * 16 + l16;
  int m = mt * 16 + l16;
  v8f acc = {};
  for (int k = 0; k < Kp; k += 4) {
    v2f a, b;
    int ka = k + 2 * half;
    a.x = A[(size_t)m * lda + ka];       // A zero-padded to Kp columns
    a.y = A[(size_t)m * lda + ka + 1];
    int kb0 = k + half, kb1 = k + 2 + half;
    b.x = (kb0 < K) ? W[(size_t)kb0 * Nn + n] : 0.f;
    b.y = (kb1 < K) ? W[(size_t)kb1 * Nn + n] : 0.f;
    acc = wmma_f32(a, b, acc);
  }
  float bs = bias[n];
#pragma unroll
  for (int i = 0; i < 8; ++i) {
    int mo = mt * 16 + i + 8 * half;
    float v = acc[i] + bs;
    if (relu) v = fmaxf(v, 0.f);
    C[(size_t)mo * ldc + n] = v;
  }
}

// ---------------- final 256 -> 2 projection ----------------

__global__ void k_out(const float* __restrict__ x, const float* __restrict__ W,
                      const float* __restrict__ b, float* __restrict__ out, int G) {
  int i = blockIdx.x * blockDim.x + threadIdx.x;
  int g = i >> 1, j = i & 1;
  if (g >= G) return;
  float s = b[j];
  for (int k = 0; k < 256; ++k) s += x[(size_t)g * 256 + k] * W[k * 2 + j];
  out[i] = s;
}

// ---------------- host orchestration ----------------

static inline unsigned cdiv(size_t a, size_t b) { return (unsigned)((a + b - 1) / b); }

extern "C" void kernel_launch(void* const* d_in, const int* in_sizes, int n_in,
                              void* d_out, int out_size, void* d_ws, size_t ws_size,
                              hipStream_t stream) {
  const float* x_a = (const float*)d_in[0];
  const int* ei_a  = (const int*)d_in[1];
  const int* bat_a = (const int*)d_in[2];
  const float* x_b = (const float*)d_in[3];
  const int* ei_b  = (const int*)d_in[4];
  const int* bat_b = (const int*)d_in[5];
  const float* W1s = (const float*)d_in[6];
  const float* b1s = (const float*)d_in[7];
  const float* W2s = (const float*)d_in[8];
  const float* b2s = (const float*)d_in[9];
  const float* gammas = (const float*)d_in[10];
  const float* betas  = (const float*)d_in[11];
  const float* fc1xd_W = (const float*)d_in[12];
  const float* fc1xd_b = (const float*)d_in[13];
  const float* fc1_W = (const float*)d_in[14];
  const float* fc1_b = (const float*)d_in[15];
  const float* fc2_W = (const float*)d_in[16];
  const float* fc2_b = (const float*)d_in[17];
  const float* out_W = (const float*)d_in[18];
  const float* out_b = (const float*)d_in[19];

  const int N = in_sizes[0] / DIMF;
  const int E = in_sizes[1] / 2;
  const int G = out_size / 2;
  const size_t NP = (size_t)N * PAD;

  float* ws = (float*)d_ws;
  size_t o = 0;
  float* xp   = ws + o; o += NP;              // padded node features (current x)
  float* agg  = ws + o; o += NP;              // neighbor aggregate
  float* hbuf = ws + o; o += NP;              // relu(MLP) pre-BN
  float* pooled = ws + o; o += (size_t)G * PAD;
  float* ha = ws + o; o += (size_t)G * 256;
  float* hb = ws + o; o += (size_t)G * 256;
  float* stats = ws + o; o += 64;
  float* meaninv = ws + o; o += 64;
  // FC head aliases the (dead after pooling) per-node buffers
  float* xc  = xp;    // G*256
  float* xc1 = agg;   // G*1024
  float* xc2 = hbuf;  // G*256

  const size_t tiles_mlp = cdiv((size_t)N, 16);

  for (int br = 0; br < 2; ++br) {
    const float* x  = br ? x_b : x_a;
    const int* ei   = br ? ei_b : ei_a;
    const int* bat  = br ? bat_b : bat_a;
    float* hout     = br ? hb : ha;

    k_pad_copy<<<cdiv(NP, 256), 256, 0, stream>>>(x, xp, N);

    for (int l = 0; l < 5; ++l) {
      k_zero<<<4096, 256, 0, stream>>>(agg, NP);
      k_scatter<<<cdiv((size_t)E * 32, 256), 256, 0, stream>>>(ei, ei + E, xp, agg, E);
      k_zero<<<1, 64, 0, stream>>>(stats, 64);
      k_gin_mlp<<<cdiv(tiles_mlp, 4), 128, 0, stream>>>(
          xp, agg, hbuf, W1s + l * DIMF * DIMF, b1s + l * DIMF,
          W2s + l * DIMF * DIMF, b2s + l * DIMF, stats, N);
      k_bn_finalize<<<1, 32, 0, stream>>>(stats, meaninv, 1.f / (float)N);
      k_bn_apply<<<cdiv(NP, 256), 256, 0, stream>>>(
          hbuf, meaninv, gammas + l * DIMF, betas + l * DIMF, xp, NP);
    }

    k_zero<<<cdiv((size_t)G * PAD, 256), 256, 0, stream>>>(pooled, (size_t)G * PAD);
    k_pool<<<cdiv(NP, 256), 256, 0, stream>>>(xp, bat, pooled, N);
    // ha/hb = relu(pooled @ fc1xd_W + b) : [G,31(pad32)] x [31,256]
    {
      size_t tiles = (size_t)(G / 16) * (256 / 16);
      k_gemm<<<cdiv(tiles, 4), 128, 0, stream>>>(pooled, PAD, fc1xd_W, fc1xd_b,
                                                 hout, 256, G, 256, DIMF, PAD, 1);
    }
  }

  k_add<<<cdiv((size_t)G * 256, 256), 256, 0, stream>>>(ha, hb, xc, (size_t)G * 256);
  {
    size_t tiles = (size_t)(G / 16) * (1024 / 16);
    k_gemm<<<cdiv(tiles, 4), 128, 0, stream>>>(xc, 256, fc1_W, fc1_b,
                                               xc1, 1024, G, 1024, 256, 256, 1);
  }
  {
    size_t tiles = (size_t)(G / 16) * (256 / 16);
    k_gemm<<<cdiv(tiles, 4), 128, 0, stream>>>(xc1, 1024, fc2_W, fc2_b,
                                               xc2, 256, G, 256, 1024, 1024, 1);
  }
  k_out<<<cdiv((size_t)G * 2, 256), 256, 0, stream>>>(xc2, out_W, out_b, (float*)d_out, G);
}